// GCNLayer_31903017074707
// MI455X (gfx1250) — compile-verified
//
#include <hip/hip_runtime.h>

typedef float v2f __attribute__((ext_vector_type(2)));
typedef float v8f __attribute__((ext_vector_type(8)));

#define D 128          // D_IN == D_OUT == 128
#define BN_EPS 1e-5f

// ---------------------------------------------------------------------------
// 1) deg init (self-loop contributes 1) + zero the BN accumulators
__global__ __launch_bounds__(256) void k_deg_init(float* deg, float* sum, float* sumsq, int N) {
    int i = blockIdx.x * blockDim.x + threadIdx.x;
    if (i < N) deg[i] = 1.0f;
    if (i < D) { sum[i] = 0.0f; sumsq[i] = 0.0f; }
}

// 2) deg[col[e]] += 1
__global__ __launch_bounds__(256) void k_deg_count(const long long* __restrict__ col,
                                                   float* __restrict__ deg, int E) {
    int e = blockIdx.x * blockDim.x + threadIdx.x;
    if (e < E) atomicAdd(&deg[(size_t)col[e]], 1.0f);
}

// 3) dinv = rsqrt(deg)   (deg >= 1 always, self-loop)
__global__ __launch_bounds__(256) void k_dinv(const float* __restrict__ deg,
                                              float* __restrict__ dinv, int N) {
    int i = blockIdx.x * blockDim.x + threadIdx.x;
    if (i < N) dinv[i] = rsqrtf(deg[i]);
}

// ---------------------------------------------------------------------------
// 4) h = x @ W^T  via fp32 WMMA 16x16x4. One wave: 16 rows x 128 cols (8 tiles).
//    A 16x4 frag: lanes 0-15 hold M=lane,(k0,k1); lanes 16-31 hold M=lane-16,(k2,k3).
//    B 4x16 frag: lanes 0-15 hold N=lane,(k0,k1); lanes 16-31 hold N=lane-16,(k2,k3).
//    D 16x16: VGPR r -> M = r + (lane<16 ? 0 : 8), N = lane&15.
__global__ __launch_bounds__(256) void k_gemm(const float* __restrict__ x,
                                              const float* __restrict__ W,
                                              float* __restrict__ h, int N) {
    int wave = (int)((blockIdx.x * blockDim.x + threadIdx.x) >> 5);
    int lane = threadIdx.x & 31;
    int m0 = wave * 16;
    if (m0 >= N) return;                         // wave-uniform exit: EXEC stays all-1s

    int half = lane >> 4;                        // 0 or 1
    int l15  = lane & 15;
    int kOff = half * 2;                         // A/B k sub-offset (0 or 2)
    const float* xrow = x + (size_t)(m0 + l15) * D + kOff;

    v8f acc[8] = {};
#if __has_builtin(__builtin_amdgcn_wmma_f32_16x16x4_f32)
    for (int k = 0; k < D; k += 4) {
        v2f a = *(const v2f*)(xrow + k);
#pragma unroll
        for (int t = 0; t < 8; ++t) {
            v2f bv = *(const v2f*)(W + (size_t)(t * 16 + l15) * D + kOff + k);
            acc[t] = __builtin_amdgcn_wmma_f32_16x16x4_f32(
                false, a, false, bv, (short)0, acc[t], false, false);
        }
    }
    int mBase = m0 + half * 8;
#pragma unroll
    for (int t = 0; t < 8; ++t)
#pragma unroll
        for (int r = 0; r < 8; ++r)
            h[(size_t)(mBase + r) * D + t * 16 + l15] = acc[t][r];
#else
    // Fallback (should not be hit on gfx1250): plain FMA, same output mapping.
    for (int n = 0; n < 8; ++n) {
        int row = m0 + half * 8;                 // unused fallback, keep simple per-lane GEMV
        (void)row;
    }
    for (int mm = 0; mm < 16; ++mm) {
        for (int nn = lane; nn < D; nn += 32) {
            float s = 0.f;
            for (int k = 0; k < D; ++k) s += x[(size_t)(m0 + mm) * D + k] * W[(size_t)nn * D + k];
            h[(size_t)(m0 + mm) * D + nn] = s;
        }
    }
#endif
}

// ---------------------------------------------------------------------------
// 5) agg[i] = h[i] * dinv[i]^2   (self-loop message; plain store -> no zero-fill)
__global__ __launch_bounds__(256) void k_agg_init(const float* __restrict__ h,
                                                  const float* __restrict__ dinv,
                                                  float* __restrict__ agg, int total4) {
    int t = blockIdx.x * blockDim.x + threadIdx.x;   // over N*32 float4s
    if (t >= total4) return;
    int i = t >> 5;
    float d = dinv[i], n = d * d;
    float4 v = ((const float4*)h)[t];
    v.x *= n; v.y *= n; v.z *= n; v.w *= n;
    ((float4*)agg)[t] = v;
}

// 6) one wave per edge: gather h[row] (float4/lane), scatter-add to agg[col] in L2
__global__ __launch_bounds__(256) void k_edge(const long long* __restrict__ row,
                                              const long long* __restrict__ col,
                                              const float* __restrict__ dinv,
                                              const float* __restrict__ h,
                                              float* __restrict__ agg, int E) {
    int e = (int)((blockIdx.x * blockDim.x + threadIdx.x) >> 5);
    if (e >= E) return;
    int lane = threadIdx.x & 31;
    long long r = row[e], c = col[e];
    float norm = dinv[(size_t)r] * dinv[(size_t)c];
    float4 v = ((const float4*)(h + (size_t)r * D))[lane];
    float* dst = agg + (size_t)c * D + lane * 4;
    atomicAdd(dst + 0, v.x * norm);
    atomicAdd(dst + 1, v.y * norm);
    atomicAdd(dst + 2, v.z * norm);
    atomicAdd(dst + 3, v.w * norm);
}

// ---------------------------------------------------------------------------
// 7) column sums / sums-of-squares of (agg + b): 128 threads = 128 features,
//    each block covers 64 rows, then 2 atomics per thread.
#define STAT_ROWS 64
__global__ __launch_bounds__(128) void k_stats(const float* __restrict__ agg,
                                               const float* __restrict__ b,
                                               float* __restrict__ sum,
                                               float* __restrict__ sumsq, int N) {
    int f  = threadIdx.x;
    int r0 = blockIdx.x * STAT_ROWS;
    int r1 = r0 + STAT_ROWS; if (r1 > N) r1 = N;
    float bb = b[f], s = 0.f, sq = 0.f;
    for (int r = r0; r < r1; ++r) {
        float v = agg[(size_t)r * D + f] + bb;
        s += v; sq += v * v;
    }
    atomicAdd(&sum[f], s);
    atomicAdd(&sumsq[f], sq);
}

// 8) fold mean/var/gamma/beta/bias into per-feature affine:  out = agg*a + b2
__global__ void k_bnprep(const float* __restrict__ sum, const float* __restrict__ sumsq,
                         const float* __restrict__ b, const float* __restrict__ gamma,
                         const float* __restrict__ beta,
                         float* __restrict__ sA, float* __restrict__ sB, float invN) {
    int f = threadIdx.x;
    float mean = sum[f] * invN;
    float var  = sumsq[f] * invN - mean * mean;
    float a    = gamma[f] * rsqrtf(var + BN_EPS);
    sA[f] = a;
    sB[f] = beta[f] + (b[f] - mean) * a;   // (agg+b-mean)*a + beta == agg*a + sB
}

// 9) in-place: out = relu(agg * a + b2), float4
__global__ __launch_bounds__(256) void k_final(float* __restrict__ out,
                                               const float* __restrict__ sA,
                                               const float* __restrict__ sB, int total4) {
    int t = blockIdx.x * blockDim.x + threadIdx.x;
    if (t >= total4) return;
    int f = (t & 31) * 4;
    float4 v = ((float4*)out)[t];
    v.x = fmaxf(v.x * sA[f + 0] + sB[f + 0], 0.f);
    v.y = fmaxf(v.y * sA[f + 1] + sB[f + 1], 0.f);
    v.z = fmaxf(v.z * sA[f + 2] + sB[f + 2], 0.f);
    v.w = fmaxf(v.w * sA[f + 3] + sB[f + 3], 0.f);
    ((float4*)out)[t] = v;
}

// ---------------------------------------------------------------------------
extern "C" void kernel_launch(void* const* d_in, const int* in_sizes, int n_in,
                              void* d_out, int out_size, void* d_ws, size_t ws_size,
                              hipStream_t stream) {
    const float*     x     = (const float*)d_in[0];
    const long long* ei    = (const long long*)d_in[1];   // int64 [2, E]
    const float*     W     = (const float*)d_in[2];
    const float*     b     = (const float*)d_in[3];
    const float*     gamma = (const float*)d_in[4];
    const float*     beta  = (const float*)d_in[5];

    const int N = in_sizes[0] / D;       // 50000
    const int E = in_sizes[1] / 2;       // 800000
    const long long* rowp = ei;
    const long long* colp = ei + E;

    float* out = (float*)d_out;

    // workspace carve-up (all fully rewritten every call)
    float* ws    = (float*)d_ws;
    float* h     = ws;                       // N*D
    float* deg   = h + (size_t)N * D;        // N
    float* dinv  = deg + N;                  // N
    float* sum   = dinv + N;                 // D
    float* sumsq = sum + D;                  // D
    float* sA    = sumsq + D;                // D
    float* sB    = sA + D;                   // D

    const int total4 = N * (D / 4);

    k_deg_init <<<(N + 255) / 256, 256, 0, stream>>>(deg, sum, sumsq, N);
    k_deg_count<<<(E + 255) / 256, 256, 0, stream>>>(colp, deg, E);
    k_dinv     <<<(N + 255) / 256, 256, 0, stream>>>(deg, dinv, N);

    int gemmWaves = (N + 15) / 16;                       // 3125 (N divisible by 16)
    k_gemm     <<<(gemmWaves + 7) / 8, 256, 0, stream>>>(x, W, h, N);

    k_agg_init <<<(total4 + 255) / 256, 256, 0, stream>>>(h, dinv, out, total4);
    k_edge     <<<(E + 7) / 8, 256, 0, stream>>>(rowp, colp, dinv, h, out, E);

    k_stats    <<<(N + STAT_ROWS - 1) / STAT_ROWS, 128, 0, stream>>>(out, b, sum, sumsq, N);
    k_bnprep   <<<1, 128, 0, stream>>>(sum, sumsq, b, gamma, beta, sA, sB, 1.0f / (float)N);
    k_final    <<<(total4 + 255) / 256, 256, 0, stream>>>(out, sA, sB, total4);
}